// FieldAwareFactorizationMachineModel_52553219834076
// MI455X (gfx1250) — compile-verified
//
#include <hip/hip_runtime.h>
#include <hip/hip_bf16.h>
#include <stdint.h>

#define NUM_FIELDS 24
#define EMBED_DIM  32
#define VOCAB      96000
#define NPAIRS     276      // 24*23/2
#define NTILES     18       // ceil(276/16)
#define WAVES      4
#define BLOCK      (WAVES * 32)

typedef __attribute__((ext_vector_type(2))) float v2f;
typedef __attribute__((ext_vector_type(8))) float v8f;

__device__ __forceinline__ uint32_t lds_addr_u32(const void* p) {
  // addrspace(3) offsets live in the low 32 bits of the generic shared address
  return (uint32_t)(uintptr_t)p;
}

__global__ __launch_bounds__(BLOCK) void ffm_kernel(
    const long long* __restrict__ x,        // [B, 24] int64
    const long long* __restrict__ offsets,  // [24]    int64
    const float*     __restrict__ ffm_W,    // [24, 96000, 32] f32
    const float*     __restrict__ fc_W,     // [96000, 1] f32
    const float*     __restrict__ bias,     // [1] f32
    float*           __restrict__ out)      // [B] f32
{
  // per-wave double-buffered tile: 32 rows x 32 floats = 4 KB per buffer
  __shared__ __align__(128) float s_tiles[WAVES][2][32 * EMBED_DIM];
  __shared__ int   s_pairs[NPAIRS];
  __shared__ int   s_ids[NUM_FIELDS];
  __shared__ float s_total;

  const int b    = blockIdx.x;
  const int tid  = threadIdx.x;
  const int lane = tid & 31;
  const int wave = tid >> 5;

  if (tid == 0) s_total = bias[0];
  if (tid < NUM_FIELDS) {
    long long g = x[(long long)b * NUM_FIELDS + tid] + offsets[tid];
    s_ids[tid] = (int)g;
  }
  // unrank upper-triangle pairs: row i (i<j) has (23-i) pairs
  for (int p = tid; p < NPAIRS; p += BLOCK) {
    int i = 0, rem = p;
    while (rem >= (NUM_FIELDS - 1) - i) { rem -= (NUM_FIELDS - 1) - i; ++i; }
    int j = i + 1 + rem;
    s_pairs[p] = (i << 8) | j;
  }
  __syncthreads();

  // linear term: sum_f fc_W[g_f]
  if (tid < NUM_FIELDS) atomicAdd(&s_total, fc_W[s_ids[tid]]);

  // ---- async gather of one 16-pair tile into LDS ----
  // slot L (0..15)  = a-row of pair q=L      : ffm_W[j, g_i, :]
  // slot L (16..31) = b-row of pair q=L-16   : ffm_W[i, g_j, :]
  auto issue_tile = [&](int t, int buf) {
    int q  = lane & 15;
    int p  = 16 * t + q;
    int pr = s_pairs[p < NPAIRS ? p : 0];   // dummy slots load pair 0 (masked later)
    int i  = pr >> 8, j = pr & 255;
    int tab, id;
    if (lane < 16) { tab = j; id = s_ids[i]; }
    else           { tab = i; id = s_ids[j]; }
    const float* row = ffm_W + (((size_t)tab * VOCAB + (size_t)id) << 5);
    uint32_t lds = lds_addr_u32(&s_tiles[wave][buf][0]) + (uint32_t)lane * 128u;
    unsigned long long ga = (unsigned long long)(uintptr_t)row;
#pragma unroll
    for (int o = 0; o < 8; ++o) {
      asm volatile("global_load_async_to_lds_b128 %0, %1, off"
                   :: "v"(lds + 16u * (unsigned)o),
                      "v"(ga + 16ull * (unsigned long long)o)
                   : "memory");
    }
  };

  float acc = 0.0f;
  int t = wave, buf = 0;
  if (t < NTILES) issue_tile(t, buf);
  while (t < NTILES) {
    int tn = t + WAVES;
    if (tn < NTILES) {
      issue_tile(tn, buf ^ 1);                              // prefetch next tile
      asm volatile("s_wait_asynccnt 0x8" ::: "memory");     // current tile done (in-order)
    } else {
      asm volatile("s_wait_asynccnt 0x0" ::: "memory");
    }

    const float* tile = &s_tiles[wave][buf][0];
    const int q  = lane & 15;
    const int hi = lane >> 4;   // lanes 16-31 carry K=4s+2,4s+3 halves of A and B

    v8f c = {};
#pragma unroll
    for (int s = 0; s < 8; ++s) {   // chain 8x K=4 WMMAs -> K=32
      const float2 av = *(const float2*)&tile[q * EMBED_DIM + 4 * s + 2 * hi];
      const float2 bv = *(const float2*)&tile[512 + q * EMBED_DIM + 4 * s + 2 * hi];
      v2f a;  a.x  = av.x; a.y  = av.y;
      v2f bb; bb.x = bv.x; bb.y = bv.y;
      c = __builtin_amdgcn_wmma_f32_16x16x4_f32(
              /*neg_a=*/false, a, /*neg_b=*/false, bb,
              /*c_mod=*/(short)0, c, /*reuse_a=*/false, /*reuse_b=*/false);
    }

    // diagonal C[p][p]: VGPR (p&7), lane p (p<8) or lane p+16 (p>=8)
    int slot = (lane < 8) ? lane : ((lane >= 24) ? (lane - 16) : -1);
    if (slot >= 0 && (16 * t + slot) < NPAIRS) {
      float d = 0.0f;
#pragma unroll
      for (int v = 0; v < 8; ++v)
        if ((slot & 7) == v) d = c[v];
      acc += d;
    }

    t = tn; buf ^= 1;
  }

  // wave32 reduce, then block accumulate
#pragma unroll
  for (int off = 16; off > 0; off >>= 1)
    acc += __shfl_xor(acc, off, 32);
  if (lane == 0) atomicAdd(&s_total, acc);

  __syncthreads();
  if (tid == 0) out[b] = 1.0f / (1.0f + expf(-s_total));
}

extern "C" void kernel_launch(void* const* d_in, const int* in_sizes, int n_in,
                              void* d_out, int out_size, void* d_ws, size_t ws_size,
                              hipStream_t stream) {
  (void)n_in; (void)out_size; (void)d_ws; (void)ws_size;
  const long long* x       = (const long long*)d_in[0];
  const long long* offsets = (const long long*)d_in[1];
  const float*     ffm_W   = (const float*)d_in[2];
  const float*     fc_W    = (const float*)d_in[3];
  const float*     bias    = (const float*)d_in[4];
  float*           out     = (float*)d_out;
  const int B = in_sizes[0] / NUM_FIELDS;
  ffm_kernel<<<B, BLOCK, 0, stream>>>(x, offsets, ffm_W, fc_W, bias, out);
}